// Label_Quantizer_23407571763539
// MI455X (gfx1250) — compile-verified
//
#include <hip/hip_runtime.h>
#include <hip/hip_bf16.h>
#include <math.h>

#define SEQ      160
#define BATCH    4096
#define NCODE    16
#define DSTATE   48
#define NBLOCKS  512   // BATCH / 8 rows per block

typedef __attribute__((ext_vector_type(2))) float v2f;
typedef __attribute__((ext_vector_type(8))) float v8f;

__device__ __forceinline__ float siluf(float v) { return v / (1.0f + expf(-v)); }
__device__ __forceinline__ float geluf(float v) { return 0.5f * v * (1.0f + erff(v * 0.70710678118654752f)); }
__device__ __forceinline__ float softplusf(float v) { return v > 20.0f ? v : log1pf(expf(v)); }

// ---------------------------------------------------------------------------
// Kernel A: batch-independent mamba branches -> m[t], t in [0,160)
// 4 waves: (branch, d) = (wave>>1, wave&1). Lanes own states s=lane, s=lane+32.
// ---------------------------------------------------------------------------
__global__ __launch_bounds__(192) void mamba_const_kernel(
    const float* __restrict__ ln_b, const float* __restrict__ in_proj,
    const float* __restrict__ out_proj,
    const float* __restrict__ cwF, const float* __restrict__ cbF,
    const float* __restrict__ xpF, const float* __restrict__ dtwF,
    const float* __restrict__ dtbF, const float* __restrict__ AlF,
    const float* __restrict__ DF,
    const float* __restrict__ cwB, const float* __restrict__ cbB,
    const float* __restrict__ xpB, const float* __restrict__ dtwB,
    const float* __restrict__ dtbB, const float* __restrict__ AlB,
    const float* __restrict__ DB,
    float* __restrict__ m_out)
{
    __shared__ float yws[4][SEQ];
    const int tid = threadIdx.x;
    const int wave = tid >> 5;
    const int lane = tid & 31;

    if (wave < 4) {
        const int branch = wave >> 1;
        const int d      = wave & 1;
        const float* cw  = branch ? cwB  : cwF;
        const float* cb  = branch ? cbB  : cbF;
        const float* xp  = branch ? xpB  : xpF;
        const float* dtw = branch ? dtwB : dtwF;
        const float* dtb = branch ? dtbB : dtbF;
        const float* Al  = branch ? AlB  : AlF;
        const float* Dp  = branch ? DB   : DF;

        const float lnb = ln_b[0];
        const float xi0 = lnb * in_proj[0];
        const float xi1 = lnb * in_proj[1];
        const float silu_z = siluf(lnb * in_proj[2 + d]);

        // depthwise causal conv of constant input: partial tap sums S[j] = sum_{k=3-j..3} w[k]
        const float w00 = cw[0], w01 = cw[1], w02 = cw[2], w03 = cw[3];
        const float w10 = cw[4], w11 = cw[5], w12 = cw[6], w13 = cw[7];
        const float cb0 = cb[0], cb1 = cb[1];
        const float S00 = w03, S01 = w02 + w03, S02 = w01 + w02 + w03, S03 = w00 + w01 + w02 + w03;
        const float S10 = w13, S11 = w12 + w13, S12 = w11 + w12 + w13, S13 = w10 + w11 + w12 + w13;

        const int s0 = lane, s1 = lane + 32;
        const bool has1 = (lane < 16);
        const float xB0a = xp[(1 + s0) * 2 + 0], xB0b = xp[(1 + s0) * 2 + 1];
        const float xC0a = xp[(1 + DSTATE + s0) * 2 + 0], xC0b = xp[(1 + DSTATE + s0) * 2 + 1];
        const float A0 = -expf(Al[d * DSTATE + s0]);
        float xB1a = 0.f, xB1b = 0.f, xC1a = 0.f, xC1b = 0.f, A1 = 0.f;
        if (has1) {
            xB1a = xp[(1 + s1) * 2 + 0]; xB1b = xp[(1 + s1) * 2 + 1];
            xC1a = xp[(1 + DSTATE + s1) * 2 + 0]; xC1b = xp[(1 + DSTATE + s1) * 2 + 1];
            A1 = -expf(Al[d * DSTATE + s1]);
        }
        const float xpd0 = xp[0], xpd1 = xp[1];
        const float dtww = dtw[d], dtbb = dtb[d], Dd = Dp[d];

        float h0 = 0.f, h1 = 0.f;
        for (int t = 0; t < SEQ; ++t) {
            const float Sa = (t >= 3) ? S03 : (t == 2 ? S02 : (t == 1 ? S01 : S00));
            const float Sb = (t >= 3) ? S13 : (t == 2 ? S12 : (t == 1 ? S11 : S10));
            const float xc0 = siluf(cb0 + xi0 * Sa);
            const float xc1 = siluf(cb1 + xi1 * Sb);
            const float xcd = d ? xc1 : xc0;
            const float dtr = xc0 * xpd0 + xc1 * xpd1;
            const float dt  = softplusf(dtr * dtww + dtbb);
            const float gin = dt * xcd;
            h0 = expf(dt * A0) * h0 + gin * (xc0 * xB0a + xc1 * xB0b);
            float part = h0 * (xc0 * xC0a + xc1 * xC0b);
            if (has1) {
                h1 = expf(dt * A1) * h1 + gin * (xc0 * xB1a + xc1 * xB1b);
                part += h1 * (xc0 * xC1a + xc1 * xC1b);
            }
            #pragma unroll
            for (int off = 16; off >= 1; off >>= 1) part += __shfl_xor(part, off, 32);
            if (lane == 0) yws[wave][t] = (part + xcd * Dd) * silu_z;
        }
    }
    __syncthreads();
    if (tid < SEQ) {
        const float op0 = out_proj[0], op1 = out_proj[1];
        // backward branch output is time-reversed at the end
        m_out[tid] = op0 * yws[0][tid] + op1 * yws[1][tid]
                   + op0 * yws[2][SEQ - 1 - tid] + op1 * yws[3][SEQ - 1 - tid];
    }
}

// ---------------------------------------------------------------------------
// Kernel B: fused dilated-conv stack + residual + pos + m[t], then WMMA
// vector-quantization. One wave per batch row, 8 rows per block.
// ---------------------------------------------------------------------------
__global__ __launch_bounds__(256) void fused_main_kernel(
    const float* __restrict__ x,
    const float* cw1, const float* cb1, const float* cw2, const float* cb2,
    const float* cw3, const float* cb3, const float* cw4, const float* cb4,
    const float* cw5, const float* cb5,
    const float* __restrict__ pos_emb, const float* __restrict__ codebook,
    const float* __restrict__ m_ws,
    float* __restrict__ out, float* __restrict__ partials)
{
    __shared__ float xorig[8][SEQ];
    __shared__ float bufA[8][SEQ];
    __shared__ float bufB[8][SEQ];
    __shared__ float code_sh[NCODE];
    __shared__ float wsum[8];

    const int tid  = threadIdx.x;
    const int wave = tid >> 5;
    const int lane = tid & 31;
    const int row  = blockIdx.x * 8 + wave;

    if (tid < NCODE) code_sh[tid] = codebook[tid];

    const float* xr = x + (size_t)row * SEQ;
    #pragma unroll
    for (int j = 0; j < 5; ++j) {
        const int t = lane + 32 * j;
        const float v = xr[t];
        xorig[wave][t] = v;
        bufA[wave][t]  = v;
    }
    __syncthreads();

    const float* cwp[5] = {cw1, cw2, cw3, cw4, cw5};
    const float* cbp[5] = {cb1, cb2, cb3, cb4, cb5};
    float* srcp = &bufA[wave][0];
    float* dstp = &bufB[wave][0];

    for (int L = 0; L < 5; ++L) {
        const int dil = 1 << L;
        const float* w = cwp[L];
        const float w0 = w[0], w1 = w[1], w2 = w[2], w3 = w[3], w4 = w[4];
        const float bb = cbp[L][0];
        #pragma unroll
        for (int j = 0; j < 5; ++j) {
            const int t = lane + 32 * j;
            const int tm2 = t - 2 * dil, tm1 = t - dil, tp1 = t + dil, tp2 = t + 2 * dil;
            // branchless 'same' padding: clamp address, mask weight
            const float vm2 = srcp[tm2 < 0 ? 0 : tm2];
            const float vm1 = srcp[tm1 < 0 ? 0 : tm1];
            const float vp1 = srcp[tp1 >= SEQ ? (SEQ - 1) : tp1];
            const float vp2 = srcp[tp2 >= SEQ ? (SEQ - 1) : tp2];
            float acc = bb + w2 * srcp[t];
            acc += (tm2 >= 0  ? w0 : 0.0f) * vm2;
            acc += (tm1 >= 0  ? w1 : 0.0f) * vm1;
            acc += (tp1 < SEQ ? w3 : 0.0f) * vp1;
            acc += (tp2 < SEQ ? w4 : 0.0f) * vp2;
            if (L < 4) acc = geluf(acc);
            dstp[t] = acc;
        }
        __syncthreads();
        float* tmp = srcp; srcp = dstp; dstp = tmp;
    }

    // enc = conv_out + x + pos_emb + m[t]  (srcp = final conv output)
    #pragma unroll
    for (int j = 0; j < 5; ++j) {
        const int t = lane + 32 * j;
        dstp[t] = srcp[t] + xorig[wave][t] + pos_emb[t] + m_ws[t];
    }
    __syncthreads();

    // --- WMMA vector quantization (codes on M, elements on N) ---
    // score[c,n] = (-2c)*enc_n + c^2*1   (same argmin over c as (enc_n - c)^2)
    // A (16x4 f32): lanes 0-15: K0 = -2*code[lane], K1 = code^2 ; lanes 16-31 (K2,K3) = 0
    // B (4x16 f32): lanes 0-15: K0 = enc[n=lane],  K1 = 1      ; lanes 16-31 (K2,K3) = 0
    // D (16x16)   : VGPR r -> code r (lanes 0-15) / code r+8 (lanes 16-31), N = lane&15
    const float* encp = dstp;
    const int  cidx  = lane & 15;
    const bool lowh  = (lane < 16);
    const int  cbase = lowh ? 0 : 8;
    const float cval = code_sh[cidx];
    v2f amat;
    amat.x = lowh ? -2.0f * cval : 0.0f;
    amat.y = lowh ? cval * cval  : 0.0f;

    float lacc = 0.0f;
    for (int m0 = 0; m0 < SEQ; m0 += 16) {
        const float ev = encp[m0 + cidx];          // enc for element n = lane&15 (all lanes)
        v2f bmat;
        bmat.x = lowh ? ev   : 0.0f;
        bmat.y = lowh ? 1.0f : 0.0f;
        v8f cz = {0.f, 0.f, 0.f, 0.f, 0.f, 0.f, 0.f, 0.f};
        v8f dsc = __builtin_amdgcn_wmma_f32_16x16x4_f32(
            false, amat, false, bmat, (short)0, cz, false, false);

        // per-lane branchless argmin over this half's 8 codes
        // (ascending r + strict '<' keeps the lowest index on ties)
        float s  = dsc[0];
        int  idx = cbase;
        #pragma unroll
        for (int r = 1; r < 8; ++r) {
            const float sr  = dsc[r];
            const bool take = sr < s;
            s   = take ? sr : s;
            idx = take ? (cbase + r) : idx;
        }
        // merge the two code halves for the same element (one exchange)
        const float so = __shfl_xor(s, 16, 32);
        const int   io = __shfl_xor(idx, 16, 32);
        const bool take = (so < s) | ((so == s) & (io < idx));
        s   = take ? so : s;
        idx = take ? io : idx;

        // both halves hold the result: low half stores q, high half stores idx
        const float q  = code_sh[idx];
        const size_t gi = (size_t)row * SEQ + (m0 + cidx);
        if (lowh) out[1 + gi] = q;                                  // q_st == q numerically
        else      out[1 + (size_t)BATCH * SEQ + gi] = (float)idx;   // idx
        const float diff = q - ev;
        lacc += lowh ? diff * diff : 0.0f;
    }

    // deterministic loss partial: fixed shuffle tree + fixed-order block sum
    #pragma unroll
    for (int off = 16; off >= 1; off >>= 1) lacc += __shfl_xor(lacc, off, 32);
    if (lane == 0) wsum[wave] = lacc;
    __syncthreads();
    if (tid == 0) {
        float s = 0.f;
        for (int i = 0; i < 8; ++i) s += wsum[i];
        partials[blockIdx.x] = s;
    }
}

// ---------------------------------------------------------------------------
// Kernel C: fixed-order reduction of block partials -> c_loss
// ---------------------------------------------------------------------------
__global__ void loss_reduce_kernel(const float* __restrict__ partials, float* __restrict__ out)
{
    if (threadIdx.x == 0 && blockIdx.x == 0) {
        float s = 0.f;
        for (int i = 0; i < NBLOCKS; ++i) s += partials[i];
        out[0] = 0.5f * s / (float)((size_t)BATCH * SEQ);
    }
}

// ---------------------------------------------------------------------------
// d_in order (setup_inputs insertion order):
//  0:x  1..10:cw1,cb1,...,cw5,cb5  11:pos_emb  12:ln_w  13:ln_b
//  14:in_proj  15:out_proj
//  16:convd_w 17:convd_b 18:xproj 19:dtproj_w 20:dtproj_b 21:A_log 22:D
//  23..29: same with _b suffix
//  30:codebook
// ---------------------------------------------------------------------------
extern "C" void kernel_launch(void* const* d_in, const int* in_sizes, int n_in,
                              void* d_out, int out_size, void* d_ws, size_t ws_size,
                              hipStream_t stream) {
    (void)in_sizes; (void)n_in; (void)out_size; (void)ws_size;
    const float* x        = (const float*)d_in[0];
    const float* cw[5]    = {(const float*)d_in[1], (const float*)d_in[3], (const float*)d_in[5],
                             (const float*)d_in[7], (const float*)d_in[9]};
    const float* cb[5]    = {(const float*)d_in[2], (const float*)d_in[4], (const float*)d_in[6],
                             (const float*)d_in[8], (const float*)d_in[10]};
    const float* pos_emb  = (const float*)d_in[11];
    const float* ln_b     = (const float*)d_in[13];
    const float* in_proj  = (const float*)d_in[14];
    const float* out_proj = (const float*)d_in[15];
    const float* codebook = (const float*)d_in[30];

    float* wsf      = (float*)d_ws;
    float* m_ws     = wsf;            // 160 floats
    float* partials = wsf + SEQ;      // 512 floats
    float* out      = (float*)d_out;

    mamba_const_kernel<<<1, 192, 0, stream>>>(
        ln_b, in_proj, out_proj,
        (const float*)d_in[16], (const float*)d_in[17], (const float*)d_in[18],
        (const float*)d_in[19], (const float*)d_in[20], (const float*)d_in[21],
        (const float*)d_in[22],
        (const float*)d_in[23], (const float*)d_in[24], (const float*)d_in[25],
        (const float*)d_in[26], (const float*)d_in[27], (const float*)d_in[28],
        (const float*)d_in[29],
        m_ws);

    fused_main_kernel<<<NBLOCKS, 256, 0, stream>>>(
        x, cw[0], cb[0], cw[1], cb[1], cw[2], cb[2], cw[3], cb[3], cw[4], cb[4],
        pos_emb, codebook, m_ws, out, partials);

    loss_reduce_kernel<<<1, 1, 0, stream>>>(partials, out);
}